// LightGCN_15006615733385
// MI455X (gfx1250) — compile-verified
//
#include <hip/hip_runtime.h>

// ---------------------------------------------------------------------------
// CDNA5 async-to-LDS path (guarded: falls back to plain loads if the builtins
// are not declared by this toolchain). Round-1 diagnostics proved both
// builtins exist on this toolchain, so USE_ASYNC_LDS == 1 here.
// ---------------------------------------------------------------------------
#if defined(__has_builtin)
#  if __has_builtin(__builtin_amdgcn_global_load_async_to_lds_b32) && \
      __has_builtin(__builtin_amdgcn_s_wait_asynccnt)
#    define USE_ASYNC_LDS 1
#  endif
#endif
#ifndef USE_ASYNC_LDS
#  define USE_ASYNC_LDS 0
#endif

#if USE_ASYNC_LDS
typedef __attribute__((address_space(1))) int as1_int;  // "global" (__device__)
typedef __attribute__((address_space(3))) int as3_int;  // "LDS"    (__shared__)

// 4-byte async copy global -> LDS. Builtin signature (from hipcc diagnostic):
// param0 = AS1 int* (global source), param1 = AS3 int* (LDS dest), imm, imm.
__device__ __forceinline__ void async_g2l_b32(const void* g, void* l) {
  int* gg = (int*)const_cast<void*>(g);   // generic, non-const, int-typed
  int* ll = (int*)l;                      // generic, int-typed
  __builtin_amdgcn_global_load_async_to_lds_b32((as1_int*)gg, (as3_int*)ll, 0, 0);
}
#endif

// ---------------------------------------------------------------------------
// Probe kernel, FIRST in the file so the disasm snippet shows its body:
// expect exactly  global_load_async_to_lds_b32 / s_wait_asynccnt /
// ds_load_b32 / global_atomic_add_f32.  Never launched; emitted anyway.
// ---------------------------------------------------------------------------
__global__ void cdna5_probe(const int* __restrict__ g, float* __restrict__ out) {
  __shared__ int s[64];
  const int t = threadIdx.x & 63;
#if USE_ASYNC_LDS
  async_g2l_b32(g + t, &s[t]);
  __builtin_amdgcn_s_wait_asynccnt(0);
#else
  s[t] = g[t];
#endif
  __syncthreads();
  atomicAdd(out + (s[t] & 1023), 1.0f);
}

#define TPB 256
#define TILE 256  // edges staged per block iteration

// ---------------------------------------------------------------------------
// SpMM scatter: embn[rows[e]+U, :] += vals[e] * embc[cols[e], :]
//
// wave32 mapping: each wave processes 2 edges per step; 16 lanes per edge,
// each lane owns 4 consecutive dims (float4 => 16 lanes * 16B = the full
// 256B embedding row, one b128 load per lane, 4 f32 atomics per lane).
// Edge metadata for a 256-edge tile is staged into LDS via the gfx1250
// async global->LDS path; the next tile is prefetched with
// global_prefetch_b8. The 128MB table + 48MB edge streams fit in 192MB L2,
// so gathers and atomic scatters are L2-resident (no NT hints on purpose).
// ---------------------------------------------------------------------------
__global__ __launch_bounds__(TPB) void spmm_scatter(
    const int* __restrict__ rows, const int* __restrict__ cols,
    const float* __restrict__ vals, const float* __restrict__ embc,
    float* __restrict__ embn, int E, int U, int numTiles) {
  __shared__ int   s_row[TILE];
  __shared__ int   s_col[TILE];
  __shared__ float s_val[TILE];

  const int t    = threadIdx.x;
  const int wave = t >> 5;        // 0..7
  const int lane = t & 31;
  const int sub  = lane >> 4;     // which edge of the pair (0/1)
  const int dl   = lane & 15;     // dim group, 4 floats each

  for (int tile = blockIdx.x; tile < numTiles; tile += gridDim.x) {
    const int base = tile * TILE;
    const int e    = base + t;

    // prefetch the tile this block will process next (gfx1250 global_prefetch_b8)
    {
      const int pe = base + (int)gridDim.x * TILE + t;
      if (pe < E) {
        __builtin_prefetch(rows + pe, 0, 1);
        __builtin_prefetch(cols + pe, 0, 1);
        __builtin_prefetch(vals + pe, 0, 1);
      }
    }

#if USE_ASYNC_LDS
    if (e < E) {
      async_g2l_b32(rows + e, &s_row[t]);
      async_g2l_b32(cols + e, &s_col[t]);
      async_g2l_b32(vals + e, &s_val[t]);
    }
    __builtin_amdgcn_s_wait_asynccnt(0);
    __syncthreads();
#else
    if (e < E) {
      s_row[t] = rows[e];
      s_col[t] = cols[e];
      s_val[t] = vals[e];
    }
    __syncthreads();
#endif

    const int remaining = min(TILE, E - base);
#pragma unroll 4
    for (int i = 0; i < 16; ++i) {
      const int slot = (wave << 5) + (i << 1) + sub;
      if (slot < remaining) {
        const int   r = s_row[slot] + U;   // item rows shifted into node space
        const int   c = s_col[slot];       // raw node index
        const float v = s_val[slot];
        const float4 x = *(const float4*)(embc + ((size_t)c << 6) + (dl << 2));
        float* dst = embn + ((size_t)r << 6) + (dl << 2);
        atomicAdd(dst + 0, v * x.x);
        atomicAdd(dst + 1, v * x.y);
        atomicAdd(dst + 2, v * x.z);
        atomicAdd(dst + 3, v * x.w);
      }
    }
    __syncthreads();  // protect LDS reuse before next tile's stores
  }
}

// ---------------------------------------------------------------------------
// utility kernels
// ---------------------------------------------------------------------------
__global__ void zero_f32(float* __restrict__ p, long n) {
  long i = (long)blockIdx.x * blockDim.x + threadIdx.x;
  if (i < n) p[i] = 0.0f;
}

__global__ void zero_f4(float4* __restrict__ p, long n4) {
  long i = (long)blockIdx.x * blockDim.x + threadIdx.x;
  if (i < n4) p[i] = make_float4(0.f, 0.f, 0.f, 0.f);
}

// row degree histogram: deg[rows[e] + U] += 1
__global__ void deg_kernel(const int* __restrict__ rows, float* __restrict__ deg,
                           int E, int U) {
  int e = blockIdx.x * blockDim.x + threadIdx.x;
  if (e < E) atomicAdd(&deg[rows[e] + U], 1.0f);
}

// in-place: deg -> deg^-0.5 (0 where deg==0), matching the reference's inf->0
__global__ void dinv_kernel(float* __restrict__ deg, int N) {
  int n = blockIdx.x * blockDim.x + threadIdx.x;
  if (n < N) {
    float d = deg[n];
    deg[n] = (d > 0.0f) ? rsqrtf(d) : 0.0f;
  }
}

// vals[e] = dinv[rows[e]+U] * dinv[cols[e]]
__global__ void vals_kernel(const int* __restrict__ rows, const int* __restrict__ cols,
                            const float* __restrict__ dinv, float* __restrict__ vals,
                            int E, int U) {
  int e = blockIdx.x * blockDim.x + threadIdx.x;
  if (e < E) vals[e] = dinv[rows[e] + U] * dinv[cols[e]];
}

// emb = concat(user_table, item_table), vectorized float4
__global__ void concat_f4(const float4* __restrict__ u, const float4* __restrict__ it,
                          float4* __restrict__ emb, long nu4, long n4) {
  long i = (long)blockIdx.x * blockDim.x + threadIdx.x;
  if (i < n4) emb[i] = (i < nu4) ? u[i] : it[i - nu4];
}

// acc += s * x   (s = 1/NUM_LAYERS, fusing the final mean)
__global__ void axpy_f4(float4* __restrict__ acc, const float4* __restrict__ x,
                        float s, long n4) {
  long i = (long)blockIdx.x * blockDim.x + threadIdx.x;
  if (i < n4) {
    float4 a = acc[i];
    float4 b = x[i];
    a.x += s * b.x; a.y += s * b.y; a.z += s * b.z; a.w += s * b.w;
    acc[i] = a;
  }
}

// ---------------------------------------------------------------------------
// host launcher
// ---------------------------------------------------------------------------
static inline size_t align256(size_t x) { return (x + 255) & ~(size_t)255; }
static inline int cdiv_i(long a, long b) { return (int)((a + b - 1) / b); }

extern "C" void kernel_launch(void* const* d_in, const int* in_sizes, int n_in,
                              void* d_out, int out_size, void* d_ws, size_t ws_size,
                              hipStream_t stream) {
  (void)n_in; (void)out_size; (void)ws_size;
  const int*   edge   = (const int*)d_in[0];   // [2, E] (rows then cols)
  const float* user_t = (const float*)d_in[1]; // [U, 64]
  const float* item_t = (const float*)d_in[2]; // [I, 64]

  const int E = in_sizes[0] / 2;
  const int U = in_sizes[1] / 64;
  const int I = in_sizes[2] / 64;
  const int N = U + I;
  const int* rows = edge;
  const int* cols = edge + E;

  // workspace carve: dinv[N] | vals[E] | embA[N*64] | embB[N*64]  (~274 MB)
  char* ws = (char*)d_ws;
  float* dinv = (float*)ws;  ws += align256((size_t)N * sizeof(float));
  float* vals = (float*)ws;  ws += align256((size_t)E * sizeof(float));
  float* embA = (float*)ws;  ws += align256((size_t)N * 64 * sizeof(float));
  float* embB = (float*)ws;
  float* acc  = (float*)d_out;

  const long n4  = (long)N * 16;      // N*64 floats as float4
  const long nu4 = (long)U * 16;

  // 1) zero accumulator (d_out is poisoned) and degree buffer
  zero_f4 <<<cdiv_i(n4, TPB), TPB, 0, stream>>>((float4*)acc, n4);
  zero_f32<<<cdiv_i(N,  TPB), TPB, 0, stream>>>(dinv, N);

  // 2) degrees -> dinv -> edge weights
  deg_kernel <<<cdiv_i(E, TPB), TPB, 0, stream>>>(rows, dinv, E, U);
  dinv_kernel<<<cdiv_i(N, TPB), TPB, 0, stream>>>(dinv, N);
  vals_kernel<<<cdiv_i(E, TPB), TPB, 0, stream>>>(rows, cols, dinv, vals, E, U);

  // 3) emb0 = concat(user, item)
  concat_f4<<<cdiv_i(n4, TPB), TPB, 0, stream>>>(
      (const float4*)user_t, (const float4*)item_t, (float4*)embA, nu4, n4);

  // 4) three propagation layers, fusing the /NUM_LAYERS mean into the add
  const int   numTiles = cdiv_i(E, TILE);
  const int   spmmGrid = numTiles < 8192 ? numTiles : 8192;
  const float scale    = 1.0f / 3.0f;
  float* cur = embA;
  float* nxt = embB;
  for (int layer = 0; layer < 3; ++layer) {
    zero_f4<<<cdiv_i(n4, TPB), TPB, 0, stream>>>((float4*)nxt, n4);
    spmm_scatter<<<spmmGrid, TPB, 0, stream>>>(rows, cols, vals, cur, nxt, E, U, numTiles);
    axpy_f4<<<cdiv_i(n4, TPB), TPB, 0, stream>>>((float4*)acc, (const float4*)nxt, scale, n4);
    float* tmp = cur; cur = nxt; nxt = tmp;
  }
}